// SAGEConvLayer_81535659147825
// MI455X (gfx1250) — compile-verified
//
#include <hip/hip_runtime.h>

typedef __attribute__((ext_vector_type(2))) float v2f;
typedef __attribute__((ext_vector_type(8))) float v8f;

#define DIM 64

// ---------------------------------------------------------------------------
// Kernel 1: zero the accumulator (d_out doubles as segment-sum buffer) and deg
// ---------------------------------------------------------------------------
__global__ void sage_init_kernel(float* __restrict__ out,
                                 float* __restrict__ deg,
                                 int n_nodes) {
  int i = blockIdx.x * blockDim.x + threadIdx.x;
  int n64 = n_nodes * DIM;
  if (i < n64) out[i] = 0.0f;
  if (i < n_nodes) deg[i] = 0.0f;
}

// ---------------------------------------------------------------------------
// Kernel 2: edge scatter. 16 threads per edge, float4 gather of x[src],
// hardware fp32 atomics into summed[dst] (L2-resident -> L2 atomic units).
// ---------------------------------------------------------------------------
__global__ void sage_scatter_kernel(const float* __restrict__ x,
                                    const long long* __restrict__ src,
                                    const long long* __restrict__ dst,
                                    float* __restrict__ summed,
                                    float* __restrict__ deg,
                                    int n_edges) {
  int gid = blockIdx.x * blockDim.x + threadIdx.x;
  int e = gid >> 4;
  int g = gid & 15;
  if (e >= n_edges) return;
  long long s = src[e];
  long long d = dst[e];
  float4 v = ((const float4*)(x + s * DIM))[g];
  float* sd = summed + d * DIM + g * 4;
  __hip_atomic_fetch_add(sd + 0, v.x, __ATOMIC_RELAXED, __HIP_MEMORY_SCOPE_AGENT);
  __hip_atomic_fetch_add(sd + 1, v.y, __ATOMIC_RELAXED, __HIP_MEMORY_SCOPE_AGENT);
  __hip_atomic_fetch_add(sd + 2, v.z, __ATOMIC_RELAXED, __HIP_MEMORY_SCOPE_AGENT);
  __hip_atomic_fetch_add(sd + 3, v.w, __ATOMIC_RELAXED, __HIP_MEMORY_SCOPE_AGENT);
  if (g == 0)
    __hip_atomic_fetch_add(deg + d, 1.0f, __ATOMIC_RELAXED, __HIP_MEMORY_SCOPE_AGENT);
}

// ---------------------------------------------------------------------------
// Kernel 3: out = (summed * 1/max(deg,1)) @ W_l + x @ W_r + b
// 8 waves/block; each wave owns a 16-row node tile; fp32 WMMA 16x16x4.
// A layout: lane L -> row m = L%16, VGPR j holds k = 2*(L/16)+j
// B layout: lane L -> col n = L%16, VGPR j holds k = 2*(L/16)+j
// C/D layout: VGPR v -> row m = v + 8*(L/16), col n = L%16
// ---------------------------------------------------------------------------
__global__ __launch_bounds__(256)
void sage_gemm_kernel(const float* __restrict__ x,
                      const float* __restrict__ Wl,
                      const float* __restrict__ Wr,
                      const float* __restrict__ bias,
                      const float* __restrict__ deg,
                      float* __restrict__ out,   // holds segment-sum on entry
                      int n_nodes) {
  __shared__ float sWl[DIM * DIM];
  __shared__ float sWr[DIM * DIM];
  for (int i = threadIdx.x; i < (DIM * DIM) / 4; i += 256) {
    ((float4*)sWl)[i] = ((const float4*)Wl)[i];
    ((float4*)sWr)[i] = ((const float4*)Wr)[i];
  }
  __syncthreads();

  int wave = threadIdx.x >> 5;
  int lane = threadIdx.x & 31;
  int n_tiles = (n_nodes + 15) >> 4;
  int tile = blockIdx.x * 8 + wave;
  if (tile >= n_tiles) return;   // wave-uniform: EXEC stays all-1s for WMMA

  int base = tile * 16;
  int m  = lane & 15;            // A-row / B-col within tile
  int kh = lane >> 4;            // which K-half this lane holds

  int arow  = base + m;
  int arowc = arow < n_nodes ? arow : (n_nodes - 1);   // clamp, keep EXEC full

  float invdeg = 1.0f / fmaxf(deg[arowc], 1.0f);

  v8f acc[4];
#pragma unroll
  for (int nt = 0; nt < 4; ++nt) {
    float bv = bias[nt * 16 + m];
#pragma unroll
    for (int v = 0; v < 8; ++v) acc[nt][v] = bv;
  }

  const float* srow = out + (long long)arowc * DIM;  // segment-sum row (in-place)
  const float* xrow = x   + (long long)arowc * DIM;

#pragma unroll
  for (int k0 = 0; k0 < DIM; k0 += 4) {
    int kk = k0 + 2 * kh;
    v2f aL, aR;
    aL.x = srow[kk]     * invdeg;
    aL.y = srow[kk + 1] * invdeg;
    aR.x = xrow[kk];
    aR.y = xrow[kk + 1];
#pragma unroll
    for (int nt = 0; nt < 4; ++nt) {
      int col = nt * 16 + m;
      v2f bL, bR;
      bL.x = sWl[kk * DIM + col];
      bL.y = sWl[(kk + 1) * DIM + col];
      bR.x = sWr[kk * DIM + col];
      bR.y = sWr[(kk + 1) * DIM + col];
      acc[nt] = __builtin_amdgcn_wmma_f32_16x16x4_f32(
          false, aL, false, bL, (short)0, acc[nt], false, false);
      acc[nt] = __builtin_amdgcn_wmma_f32_16x16x4_f32(
          false, aR, false, bR, (short)0, acc[nt], false, false);
    }
  }

  // Store D: per-lane row guard is fine here (all WMMAs are done).
#pragma unroll
  for (int nt = 0; nt < 4; ++nt) {
#pragma unroll
    for (int v = 0; v < 8; ++v) {
      int row = base + v + 8 * kh;
      if (row < n_nodes)
        out[(long long)row * DIM + nt * 16 + m] = acc[nt][v];
    }
  }
}

// ---------------------------------------------------------------------------
extern "C" void kernel_launch(void* const* d_in, const int* in_sizes, int n_in,
                              void* d_out, int out_size, void* d_ws, size_t ws_size,
                              hipStream_t stream) {
  const float*      x  = (const float*)d_in[0];
  const long long*  ei = (const long long*)d_in[1];
  const float*      Wl = (const float*)d_in[2];
  const float*      Wr = (const float*)d_in[3];
  const float*      b  = (const float*)d_in[4];
  float* out = (float*)d_out;
  float* deg = (float*)d_ws;   // n_nodes floats of scratch

  int n_nodes = in_sizes[0] / DIM;
  int n_edges = in_sizes[1] / 2;
  const long long* src = ei;
  const long long* dst = ei + n_edges;

  {
    int total  = n_nodes * DIM;
    int blocks = (total + 255) / 256;
    sage_init_kernel<<<blocks, 256, 0, stream>>>(out, deg, n_nodes);
  }
  {
    long long total = (long long)n_edges * 16;
    int blocks = (int)((total + 255) / 256);
    sage_scatter_kernel<<<blocks, 256, 0, stream>>>(x, src, dst, out, deg, n_edges);
  }
  {
    int n_tiles = (n_nodes + 15) / 16;
    int blocks  = (n_tiles + 7) / 8;
    sage_gemm_kernel<<<blocks, 256, 0, stream>>>(x, Wl, Wr, b, deg, out, n_nodes);
  }
}